// PRSGNN_36979668418675
// MI455X (gfx1250) — compile-verified
//
#include <hip/hip_runtime.h>

typedef __attribute__((ext_vector_type(2))) float v2f;
typedef __attribute__((ext_vector_type(8))) float v8f;

#define HD 16

// ---------------- elementwise helpers ----------------

__global__ __launch_bounds__(256) void k_zero(float* __restrict__ p, long long n) {
    long long i = (long long)blockIdx.x * blockDim.x + threadIdx.x;
    if (i < n) p[i] = 0.0f;
}

// deg[dst] += 1 for each edge
__global__ __launch_bounds__(256) void k_degree(const long long* __restrict__ dst,
                                                float* __restrict__ deg, int E) {
    int e = blockIdx.x * blockDim.x + threadIdx.x;
    if (e < E) atomicAdd(&deg[(int)dst[e]], 1.0f);
}

// dinv[i] = rsqrt(deg[i] + 1)   (self-loop)
__global__ __launch_bounds__(256) void k_dinv(float* __restrict__ d, int n) {
    int i = blockIdx.x * blockDim.x + threadIdx.x;
    if (i < n) d[i] = rsqrtf(d[i] + 1.0f);
}

// H[i][j] = x[i] * W1[j]   (d_in == 1, so x@W1 is an outer product)
__global__ __launch_bounds__(256) void k_h1(const float* __restrict__ x,
                                            const float* __restrict__ W1,
                                            float* __restrict__ H, int n) {
    int g = blockIdx.x * blockDim.x + threadIdx.x;
    if (g < n * HD) H[g] = x[g >> 4] * W1[g & 15];
}

// 16 threads per edge: G[dst*16+c] += H[src*16+c] * dinv[src]*dinv[dst]
__global__ __launch_bounds__(256) void k_scatter(const long long* __restrict__ src,
                                                 const long long* __restrict__ dst,
                                                 const float* __restrict__ dinv,
                                                 const float* __restrict__ H,
                                                 float* __restrict__ G, int E) {
    long long g = (long long)blockIdx.x * blockDim.x + threadIdx.x;
    if (g >= (long long)E * HD) return;
    int e = (int)(g >> 4);
    int c = (int)(g & 15);
    int s = (int)src[e];
    int d = (int)dst[e];
    float norm = dinv[s] * dinv[d];
    atomicAdd(&G[(long long)d * HD + c], H[(long long)s * HD + c] * norm);
}

// H[i][j] = relu(G[i][j] + H[i][j]*dinv[i]^2 + b[j])
__global__ __launch_bounds__(256) void k_finish_relu(const float* __restrict__ G,
                                                     float* __restrict__ H,
                                                     const float* __restrict__ dinv,
                                                     const float* __restrict__ b, int n) {
    int g = blockIdx.x * blockDim.x + threadIdx.x;
    if (g >= n * HD) return;
    int i = g >> 4;
    float di = dinv[i];
    float v = G[g] + H[g] * di * di + b[g & 15];
    H[g] = fmaxf(v, 0.0f);
}

// ---------------- WMMA GEMM: G[N,16] = H[N,16] @ W[16,16] (fp32) ----------------
// One wave per 16x16 node tile. K=16 via 4 chained V_WMMA_F32_16X16X4_F32.
// Fragment layouts per CDNA5 ISA 7.12.2:
//   A 16x4 f32 (2 VGPRs): lane&15 = M; lane>=16 -> K+2; vgpr index -> K+1
//   B 4x16  f32 (2 VGPRs): lane&15 = N; lane>=16 -> K+2; vgpr index -> K+1
//   C/D 16x16 f32 (8 VGPRs): vgpr v -> M=v (lanes 0-15) / M=v+8 (lanes 16-31); N=lane&15
__global__ __launch_bounds__(256) void k_gemm_wmma(const float* __restrict__ H,
                                                   const float* __restrict__ W,
                                                   float* __restrict__ G, int ntiles) {
    int wave = (blockIdx.x * blockDim.x + threadIdx.x) >> 5;  // wave32
    int lane = threadIdx.x & 31;
    if (wave >= ntiles) return;                // wave-uniform; EXEC all-1s inside

    int r0    = wave * HD;                     // first node row of this tile
    int mn    = lane & 15;                     // M for A, N for B/D
    int khalf = (lane >> 4) * 2;               // upper half of wave -> K offset +2
    int dhalf = (lane >> 4) * 8;               // upper half of wave -> M offset +8 in D

    v8f c = {};
    const float* arow = H + (long long)(r0 + mn) * HD;
#pragma unroll
    for (int kc = 0; kc < 4; ++kc) {
        int k0 = kc * 4 + khalf;
        v2f a, b;
        a.x = arow[k0];
        a.y = arow[k0 + 1];
        b.x = W[(k0)     * HD + mn];
        b.y = W[(k0 + 1) * HD + mn];
        // 8 args: (neg_a, A, neg_b, B, c_mod, C, reuse_a, reuse_b)
        c = __builtin_amdgcn_wmma_f32_16x16x4_f32(false, a, false, b,
                                                  (short)0, c, false, false);
    }
#pragma unroll
    for (int v = 0; v < 8; ++v)
        G[(long long)(r0 + dhalf + v) * HD + mn] = c[v];
}

// scalar tail for rows not covered by full 16-row tiles (n % 16 != 0 safety)
__global__ __launch_bounds__(256) void k_gemm_tail(const float* __restrict__ H,
                                                   const float* __restrict__ W,
                                                   float* __restrict__ G,
                                                   int row0, int n) {
    int g = blockIdx.x * blockDim.x + threadIdx.x;
    int i = row0 + (g >> 4);
    if (i >= n) return;
    int j = g & 15;
    float acc = 0.0f;
#pragma unroll
    for (int k = 0; k < HD; ++k) acc += H[(long long)i * HD + k] * W[k * HD + j];
    G[(long long)i * HD + j] = acc;
}

// out[i] = sum_j relu(AGG[i][j] + Hh[i][j]*dinv^2 + b2[j]) * Wl[j] + bl
__global__ __launch_bounds__(256) void k_final(const float* __restrict__ AGG,
                                               const float* __restrict__ Hh,
                                               const float* __restrict__ dinv,
                                               const float* __restrict__ b2,
                                               const float* __restrict__ Wl,
                                               const float* __restrict__ bl,
                                               float* __restrict__ out, int n) {
    int i = blockIdx.x * blockDim.x + threadIdx.x;
    if (i >= n) return;
    float di  = dinv[i];
    float di2 = di * di;
    float acc = bl[0];
    const float* ar = AGG + (long long)i * HD;
    const float* hr = Hh  + (long long)i * HD;
#pragma unroll
    for (int j = 0; j < HD; ++j) {
        float v = ar[j] + hr[j] * di2 + b2[j];
        acc += fmaxf(v, 0.0f) * Wl[j];
    }
    out[i] = acc;
}

// ---------------- launcher ----------------

extern "C" void kernel_launch(void* const* d_in, const int* in_sizes, int n_in,
                              void* d_out, int out_size, void* d_ws, size_t ws_size,
                              hipStream_t stream) {
    const float*     x   = (const float*)d_in[0];
    const long long* ei  = (const long long*)d_in[1];   // int64 edge_index [2,E]
    const float*     W1  = (const float*)d_in[2];
    const float*     b1  = (const float*)d_in[3];
    const float*     W2  = (const float*)d_in[4];
    const float*     b2  = (const float*)d_in[5];
    const float*     Wl  = (const float*)d_in[6];
    const float*     bl  = (const float*)d_in[7];
    float*           out = (float*)d_out;

    const int n = in_sizes[0];          // 200000 nodes
    const int E = in_sizes[1] / 2;      // 6400000 edges
    const long long* src = ei;
    const long long* dst = ei + E;

    float* ws   = (float*)d_ws;
    float* dinv = ws;                              // n floats
    float* H    = ws + n;                          // 16n floats
    float* G    = ws + n + (long long)HD * n;      // 16n floats

    const int B = 256;
    const long long nf  = (long long)n * HD;
    const long long esc = (long long)E * HD;
    dim3 gN((n + B - 1) / B), gE((E + B - 1) / B);
    dim3 gNF((unsigned)((nf + B - 1) / B)), gES((unsigned)((esc + B - 1) / B));

    // degree + symmetric norm
    k_zero  <<<gN,  B, 0, stream>>>(dinv, n);
    k_degree<<<gE,  B, 0, stream>>>(dst, dinv, E);
    k_dinv  <<<gN,  B, 0, stream>>>(dinv, n);

    // ---- layer 1: h1 = x @ W1; aggregate; relu ----
    k_h1     <<<gNF, B, 0, stream>>>(x, W1, H, n);
    k_zero   <<<gNF, B, 0, stream>>>(G, nf);
    k_scatter<<<gES, B, 0, stream>>>(src, dst, dinv, H, G, E);
    k_finish_relu<<<gNF, B, 0, stream>>>(G, H, dinv, b1, n);   // H := layer-1 activations

    // ---- layer 2: h2 = H @ W2 (WMMA fp32); aggregate ----
    int ntiles = n / HD;
    if (ntiles > 0) {
        dim3 gT((ntiles + 7) / 8);                 // 8 waves per 256-thread block
        k_gemm_wmma<<<gT, B, 0, stream>>>(H, W2, G, ntiles);
    }
    if (n & 15) {
        int rem = n - ntiles * HD;
        dim3 gR((rem * HD + B - 1) / B);
        k_gemm_tail<<<gR, B, 0, stream>>>(H, W2, G, ntiles * HD, n);
    }
    k_zero   <<<gNF, B, 0, stream>>>(H, nf);
    k_scatter<<<gES, B, 0, stream>>>(src, dst, dinv, G, H, E);

    // ---- fused: relu(agg2 + self + b2) @ Wl + bl ----
    k_final<<<gN, B, 0, stream>>>(H, G, dinv, b2, Wl, bl, out, n);
}